// Sparsemax_61349312856633
// MI455X (gfx1250) — compile-verified
//
#include <hip/hip_runtime.h>
#include <hip/hip_bf16.h>
#include <stdint.h>

// Row-wise sparsemax for a [rows x 32000] fp32 matrix (MI455X / gfx1250).
//
// Roofline: 524 MB read + 524 MB write => ~45 us at 23.3 TB/s. Pure HBM bound.
// Design:
//  - 1 workgroup (1024 thr = 32 wave32) per row.
//  - CDNA5 async DMA stage: GLOBAL_LOAD_ASYNC_TO_LDS_B128 (ASYNCcnt) pulls the
//    full 128 KB row into LDS (fits: 320 KB/WGP; 2 blocks co-resident/WGP).
//  - LDS is read exactly ONCE into registers (8 x float4 = 32 VGPR/thread
//    holds the whole row across the block) -> iterative threshold search runs
//    entirely on registers, so LDS/HBM traffic stays at the minimum.
//  - Fused max+sum pass makes Michelot iteration 0 free; subsequent
//    iterations (tau <- (sum_{z>tau} z - 1)/k) converge to the exact
//    sort-free sparsemax threshold in a handful of rounds.
//  - Output streamed with non-temporal stores (write-once, bypass caches).

#define N_COLS    32000
#define NVEC      (N_COLS / 4)     // 8000 float4 per row
#define THREADS   1024
#define NWAVES    (THREADS / 32)   // 32 waves
#define FULLJ     7                // slots 0..6 always in-range
#define PER_THR   8                // ceil(8000 / 1024)
#define MAX_ITERS 48

typedef float nvec4 __attribute__((ext_vector_type(4)));

__global__ __launch_bounds__(THREADS)
void sparsemax_rows_kernel(const float* __restrict__ x,
                           float* __restrict__ out,
                           int rows) {
    __shared__ float4 s_row[NVEC];     // 128000 B staging buffer
    __shared__ float2 s_red[NWAVES];   // cross-wave reduction scratch

    const int tid  = threadIdx.x;
    const int lane = tid & 31;
    const int wv   = tid >> 5;
    const int row  = blockIdx.x;
    if (row >= rows) return;           // block-uniform (grid == rows)

    const uint64_t gbase =
        (uint64_t)(uintptr_t)(x + (size_t)row * (size_t)N_COLS);

    // ---- Stage row: HBM -> LDS via CDNA5 async copies (ASYNCcnt) ----
    {
        const uint32_t lds0 = (uint32_t)(uintptr_t)(&s_row[0]);
#pragma unroll
        for (int j = 0; j < FULLJ; ++j) {
            uint32_t off = (uint32_t)(tid + j * THREADS) * 16u;
            uint32_t lds = lds0 + off;
            asm volatile("global_load_async_to_lds_b128 %0, %1, %2"
                         :: "v"(lds), "v"(off), "s"(gbase)
                         : "memory");
        }
        if (tid + FULLJ * THREADS < NVEC) {
            uint32_t off = (uint32_t)(tid + FULLJ * THREADS) * 16u;
            uint32_t lds = lds0 + off;
            asm volatile("global_load_async_to_lds_b128 %0, %1, %2"
                         :: "v"(lds), "v"(off), "s"(gbase)
                         : "memory");
        }
        asm volatile("s_wait_asynccnt 0" ::: "memory");
    }
    __syncthreads();

    // ---- Single LDS read pass: row -> registers ----
    const float SENT = -3.402823466e38f;   // sentinel for the ragged tail
    float4 r[PER_THR];
#pragma unroll
    for (int j = 0; j < FULLJ; ++j) r[j] = s_row[tid + j * THREADS];
    {
        const int it8 = tid + FULLJ * THREADS;
        r[FULLJ] = (it8 < NVEC) ? s_row[it8]
                                : make_float4(SENT, SENT, SENT, SENT);
    }
    const bool tail_ok = (tid + FULLJ * THREADS) < NVEC;

    // ---- Fused pass: row max + raw sum (registers only) ----
    float m = SENT, s0 = 0.0f;
#pragma unroll
    for (int j = 0; j < FULLJ; ++j) {
        float4 v = r[j];
        m  = fmaxf(m, fmaxf(fmaxf(v.x, v.y), fmaxf(v.z, v.w)));
        s0 += (v.x + v.y) + (v.z + v.w);
    }
    if (tail_ok) {
        float4 v = r[FULLJ];
        m  = fmaxf(m, fmaxf(fmaxf(v.x, v.y), fmaxf(v.z, v.w)));
        s0 += (v.x + v.y) + (v.z + v.w);
    }
#pragma unroll
    for (int o = 16; o > 0; o >>= 1) {
        m  = fmaxf(m, __shfl_xor(m, o, 32));
        s0 += __shfl_xor(s0, o, 32);
    }
    if (lane == 0) s_red[wv] = make_float2(m, s0);
    __syncthreads();
    if (wv == 0) {
        float2 w = s_red[lane];
#pragma unroll
        for (int o = 16; o > 0; o >>= 1) {
            w.x = fmaxf(w.x, __shfl_xor(w.x, o, 32));
            w.y += __shfl_xor(w.y, o, 32);
        }
        if (lane == 0) s_red[0] = w;
    }
    __syncthreads();
    const float mx      = s_red[0].x;
    const float sum_all = s_red[0].y;

    // ---- Michelot iteration on registers ----
    // Iteration 0 comes free from the fused pass: all elements active.
    float tau    = ((sum_all - (float)N_COLS * mx) - 1.0f) / (float)N_COLS;
    float prev_k = (float)N_COLS;
    for (int it = 0; it < MAX_ITERS; ++it) {
        float s = 0.0f, k = 0.0f;
#pragma unroll
        for (int j = 0; j < PER_THR; ++j) {
            float4 v = r[j];
            float z0 = v.x - mx, z1 = v.y - mx;
            float z2 = v.z - mx, z3 = v.w - mx;
            if (z0 > tau) { s += z0; k += 1.0f; }   // sentinel -> -inf: never
            if (z1 > tau) { s += z1; k += 1.0f; }   // active, adds nothing
            if (z2 > tau) { s += z2; k += 1.0f; }
            if (z3 > tau) { s += z3; k += 1.0f; }
        }
#pragma unroll
        for (int o = 16; o > 0; o >>= 1) {
            s += __shfl_xor(s, o, 32);
            k += __shfl_xor(k, o, 32);
        }
        __syncthreads();                 // protect s_red reuse across iters
        if (lane == 0) s_red[wv] = make_float2(s, k);
        __syncthreads();
        if (wv == 0) {
            float2 w = s_red[lane];
#pragma unroll
            for (int o = 16; o > 0; o >>= 1) {
                w.x += __shfl_xor(w.x, o, 32);
                w.y += __shfl_xor(w.y, o, 32);
            }
            if (lane == 0) s_red[0] = w;
        }
        __syncthreads();
        float2 t = s_red[0];             // uniform across the block
        tau = (t.x - 1.0f) / t.y;
        if (t.y == prev_k) break;        // active set stable -> tau exact
        prev_k = t.y;
    }

    // ---- Output from registers, non-temporal streaming stores ----
    const float thr = mx + tau;
    nvec4* __restrict__ gout =
        (nvec4*)(out + (size_t)row * (size_t)N_COLS);
#pragma unroll
    for (int j = 0; j < FULLJ; ++j) {
        float4 v = r[j];
        nvec4 o = { fmaxf(v.x - thr, 0.0f), fmaxf(v.y - thr, 0.0f),
                    fmaxf(v.z - thr, 0.0f), fmaxf(v.w - thr, 0.0f) };
        __builtin_nontemporal_store(o, &gout[tid + j * THREADS]);
    }
    if (tail_ok) {
        float4 v = r[FULLJ];
        nvec4 o = { fmaxf(v.x - thr, 0.0f), fmaxf(v.y - thr, 0.0f),
                    fmaxf(v.z - thr, 0.0f), fmaxf(v.w - thr, 0.0f) };
        __builtin_nontemporal_store(o, &gout[tid + FULLJ * THREADS]);
    }
}

extern "C" void kernel_launch(void* const* d_in, const int* in_sizes, int n_in,
                              void* d_out, int out_size, void* d_ws, size_t ws_size,
                              hipStream_t stream) {
    (void)n_in; (void)out_size; (void)d_ws; (void)ws_size;
    const float* x   = (const float*)d_in[0];
    float*       out = (float*)d_out;
    const int rows = in_sizes[0] / N_COLS;   // 4096 for the reference shapes
    sparsemax_rows_kernel<<<rows, THREADS, 0, stream>>>(x, out, rows);
}